// Conv_18708877541970
// MI455X (gfx1250) — compile-verified
//
#include <hip/hip_runtime.h>
#include <hip/hip_bf16.h>

typedef float v2f __attribute__((ext_vector_type(2)));
typedef float v8f __attribute__((ext_vector_type(8)));

// ---------------------------------------------------------------------------
// f32 WMMA batched GEMV:  acc += W(32x32 row-major) @ x(32)
// V_WMMA_F32_16X16X4_F32:  D(16x16) = A(16x4) * B(4x16) + C
// A layout (ISA 7.12.2): VGPR0 = K0 (lanes 0-15) | K2 (lanes 16-31), VGPR1 = K1 | K3
// B filled so that B[k][n] = x[k] for ALL n -> every D column computes W_tile@x,
// i.e. the result is replicated across the 16 column-lanes of each half.
// C/D layout: lanes 0-15 hold rows {j, 16+j}, lanes 16-31 hold rows {8+j, 24+j}.
// ---------------------------------------------------------------------------
__device__ __forceinline__ void gemv32_wmma(const float* __restrict__ W,
                                            const float* __restrict__ x,
                                            int lane, v8f& acc0, v8f& acc1) {
  const int rowA = lane & 15;
  const int koff = (lane >> 4) << 1;  // 0 for lanes 0-15, 2 for lanes 16-31
#pragma unroll
  for (int k0 = 0; k0 < 32; k0 += 4) {
    v2f b;
    b.x = x[k0 + koff];
    b.y = x[k0 + koff + 1];
    v2f a0 = *(const v2f*)(W + rowA * 32 + k0 + koff);          // rows 0..15 tile
    acc0 = __builtin_amdgcn_wmma_f32_16x16x4_f32(false, a0, false, b,
                                                 (short)0, acc0, false, false);
    v2f a1 = *(const v2f*)(W + (rowA + 16) * 32 + k0 + koff);   // rows 16..31 tile
    acc1 = __builtin_amdgcn_wmma_f32_16x16x4_f32(false, a1, false, b,
                                                 (short)0, acc1, false, false);
  }
}

// float atomic max via signed/unsigned integer ordering trick (init = -inf)
__device__ __forceinline__ void atomicMaxF(float* addr, float v) {
  if (v >= 0.0f) atomicMax((int*)addr, __float_as_int(v));
  else           atomicMin((unsigned int*)addr, (unsigned int)__float_as_int(v));
}

// ---------------------------------------------------------------------------
__global__ void init_kernel(float* mbuf, float* denom, float* agg, int N) {
  int t = blockIdx.x * blockDim.x + threadIdx.x;
  if (t < N * 4) { mbuf[t] = -INFINITY; denom[t] = 0.0f; }
  if (t < N * 32) agg[t] = 0.0f;
}

// one wave per edge
__global__ void edge_kernel(const float* __restrict__ feat,
                            const float* __restrict__ query,
                            const float* __restrict__ skw, const float* __restrict__ dkw,
                            const float* __restrict__ skb, const float* __restrict__ dkb,
                            const float* __restrict__ svw, const float* __restrict__ dvw,
                            const float* __restrict__ svb, const float* __restrict__ dvb,
                            const int* __restrict__ src, const int* __restrict__ dst,
                            float* __restrict__ key_out, float* __restrict__ value_out,
                            float* __restrict__ logits, float* __restrict__ mbuf, int E) {
  const int lane = threadIdx.x & 31;
  int e = (blockIdx.x * blockDim.x + threadIdx.x) >> 5;
  e = __builtin_amdgcn_readfirstlane(e);
  if (e >= E) return;  // wave-uniform: EXEC stays all-ones for WMMA below

  int s = src[e]; s = __builtin_amdgcn_readfirstlane(s);
  int d = dst[e]; d = __builtin_amdgcn_readfirstlane(d);
  const float* fs = feat + (long)s * 32;
  const float* fd = feat + (long)d * 32;
  const int grp   = lane >> 4;     // which lane-half
  const int rbase = grp ? 8 : 0;   // first row owned by this half in acc0

  if (e + 1 < E) {                 // gfx1250 global_prefetch_b8 of next edge's weights
    __builtin_prefetch(skw + (long)(e + 1) * 1024, 0, 1);
    __builtin_prefetch(dkw + (long)(e + 1) * 1024, 0, 1);
    __builtin_prefetch(svw + (long)(e + 1) * 1024, 0, 1);
    __builtin_prefetch(dvw + (long)(e + 1) * 1024, 0, 1);
  }

  // ------------------ KEY = relu(Wsk@fs + bsk + Wdk@fd + bdk) ------------------
  v8f a0 = {0.f,0.f,0.f,0.f,0.f,0.f,0.f,0.f};
  v8f a1 = {0.f,0.f,0.f,0.f,0.f,0.f,0.f,0.f};
  gemv32_wmma(skw + (long)e * 1024, fs, lane, a0, a1);
  gemv32_wmma(dkw + (long)e * 1024, fd, lane, a0, a1);

  const float* b1 = skb + (long)e * 32;
  const float* b2 = dkb + (long)e * 32;
  float k[16];
#pragma unroll
  for (int j = 0; j < 8; ++j) {
    const int r0 = rbase + j, r1 = rbase + 16 + j;
    k[j]     = fmaxf(a0[j] + b1[r0] + b2[r0], 0.0f);
    k[8 + j] = fmaxf(a1[j] + b1[r1] + b2[r1], 0.0f);
  }

  // logits: half 0 owns heads {0,2}, half 1 owns heads {1,3} (8 rows each)
  const float* q = query + (long)d * 32;
  float l0 = 0.0f, l1 = 0.0f;
#pragma unroll
  for (int j = 0; j < 8; ++j) {
    l0 += k[j]     * q[rbase + j];
    l1 += k[8 + j] * q[rbase + 16 + j];
  }

  if ((lane & 15) == 0) {  // lanes 0 and 16: results are replicated within each half
    float* ko = key_out + (long)e * 32 + rbase;
    *(float4*)(ko)      = make_float4(k[0],  k[1],  k[2],  k[3]);
    *(float4*)(ko + 4)  = make_float4(k[4],  k[5],  k[6],  k[7]);
    *(float4*)(ko + 16) = make_float4(k[8],  k[9],  k[10], k[11]);
    *(float4*)(ko + 20) = make_float4(k[12], k[13], k[14], k[15]);
    const int h0 = grp, h1 = grp + 2;
    logits[(long)e * 4 + h0] = l0;
    logits[(long)e * 4 + h1] = l1;
    atomicMaxF(&mbuf[(long)d * 4 + h0], l0);
    atomicMaxF(&mbuf[(long)d * 4 + h1], l1);
  }

  // ------------------ VALUE = relu(Wsv@fs + bsv + Wdv@fd + bdv) ----------------
  v8f c0 = {0.f,0.f,0.f,0.f,0.f,0.f,0.f,0.f};
  v8f c1 = {0.f,0.f,0.f,0.f,0.f,0.f,0.f,0.f};
  gemv32_wmma(svw + (long)e * 1024, fs, lane, c0, c1);
  gemv32_wmma(dvw + (long)e * 1024, fd, lane, c0, c1);

  const float* b3 = svb + (long)e * 32;
  const float* b4 = dvb + (long)e * 32;
  float v[16];
#pragma unroll
  for (int j = 0; j < 8; ++j) {
    const int r0 = rbase + j, r1 = rbase + 16 + j;
    v[j]     = fmaxf(c0[j] + b3[r0] + b4[r0], 0.0f);
    v[8 + j] = fmaxf(c1[j] + b3[r1] + b4[r1], 0.0f);
  }
  if ((lane & 15) == 0) {
    float* vo = value_out + (long)e * 32 + rbase;
    *(float4*)(vo)      = make_float4(v[0],  v[1],  v[2],  v[3]);
    *(float4*)(vo + 4)  = make_float4(v[4],  v[5],  v[6],  v[7]);
    *(float4*)(vo + 16) = make_float4(v[8],  v[9],  v[10], v[11]);
    *(float4*)(vo + 20) = make_float4(v[12], v[13], v[14], v[15]);
  }
}

// ex = exp(logit - m[dst]); denom[dst] += ex   (one thread per edge-head)
__global__ void expsum_kernel(float* __restrict__ logits, const float* __restrict__ mbuf,
                              float* __restrict__ denom, const int* __restrict__ dst, int E) {
  int t = blockIdx.x * blockDim.x + threadIdx.x;
  if (t >= E * 4) return;
  const int e = t >> 2, h = t & 3;
  const int d = dst[e];
  float ex = expf(logits[t] - mbuf[(long)d * 4 + h]);
  logits[t] = ex;  // reuse buffer (rewritten by edge_kernel every replay)
  atomicAdd(&denom[(long)d * 4 + h], ex);
}

// attn = ex/denom[dst]; agg[dst] += attn*value   (one thread per edge-element)
__global__ void attn_kernel(const float* __restrict__ ex, const float* __restrict__ denom,
                            const float* __restrict__ value_out, const int* __restrict__ dst,
                            float* __restrict__ attn_out, float* __restrict__ agg, int E) {
  int t = blockIdx.x * blockDim.x + threadIdx.x;
  if (t >= E * 32) return;
  const int e = t >> 5, i = t & 31, h = i >> 3;
  const int d = dst[e];
  const float a = ex[(long)e * 4 + h] / denom[(long)d * 4 + h];
  atomicAdd(&agg[(long)d * 32 + i], value_out[t] * a);
  if ((i & 7) == 0) attn_out[(long)e * 4 + h] = a;
}

// one wave per node: out = LN(relu(W_n @ agg_n + b_n) + feat_n)
__global__ void node_kernel(const float* __restrict__ feat, const float* __restrict__ nw,
                            const float* __restrict__ nb, const float* __restrict__ gamma,
                            const float* __restrict__ beta, const float* __restrict__ agg,
                            float* __restrict__ out, int N) {
  const int lane = threadIdx.x & 31;
  int n = (blockIdx.x * blockDim.x + threadIdx.x) >> 5;
  n = __builtin_amdgcn_readfirstlane(n);
  if (n >= N) return;

  const float* x = agg + (long)n * 32;
  v8f a0 = {0.f,0.f,0.f,0.f,0.f,0.f,0.f,0.f};
  v8f a1 = {0.f,0.f,0.f,0.f,0.f,0.f,0.f,0.f};
  gemv32_wmma(nw + (long)n * 1024, x, lane, a0, a1);

  const int grp = lane >> 4, rbase = grp ? 8 : 0;
  const float* bb = nb + (long)n * 32;
  const float* fr = feat + (long)n * 32;
  float y[16];
  float s1 = 0.0f, s2 = 0.0f;
#pragma unroll
  for (int j = 0; j < 8; ++j) {
    const int r0 = rbase + j, r1 = rbase + 16 + j;
    y[j]     = fmaxf(a0[j] + bb[r0], 0.0f) + fr[r0];
    y[8 + j] = fmaxf(a1[j] + bb[r1], 0.0f) + fr[r1];
    s1 += y[j] + y[8 + j];
    s2 += y[j] * y[j] + y[8 + j] * y[8 + j];
  }
  // each half holds 16 of 32 values (replicated across its lanes); combine halves
  s1 += __shfl_xor(s1, 16);
  s2 += __shfl_xor(s2, 16);
  const float mu  = s1 * (1.0f / 32.0f);
  const float var = s2 * (1.0f / 32.0f) - mu * mu;
  const float rs  = rsqrtf(var + 1e-5f);
  float z[16];
#pragma unroll
  for (int j = 0; j < 8; ++j) {
    const int r0 = rbase + j, r1 = rbase + 16 + j;
    z[j]     = (y[j]     - mu) * rs * gamma[r0] + beta[r0];
    z[8 + j] = (y[8 + j] - mu) * rs * gamma[r1] + beta[r1];
  }
  if ((lane & 15) == 0) {
    float* oo = out + (long)n * 32 + rbase;
    *(float4*)(oo)      = make_float4(z[0],  z[1],  z[2],  z[3]);
    *(float4*)(oo + 4)  = make_float4(z[4],  z[5],  z[6],  z[7]);
    *(float4*)(oo + 16) = make_float4(z[8],  z[9],  z[10], z[11]);
    *(float4*)(oo + 20) = make_float4(z[12], z[13], z[14], z[15]);
  }
}

// ---------------------------------------------------------------------------
extern "C" void kernel_launch(void* const* d_in, const int* in_sizes, int n_in,
                              void* d_out, int out_size, void* d_ws, size_t ws_size,
                              hipStream_t stream) {
  const float* feat   = (const float*)d_in[0];
  const float* query  = (const float*)d_in[1];
  const float* node_w = (const float*)d_in[2];
  const float* node_b = (const float*)d_in[3];
  const float* skw    = (const float*)d_in[4];
  const float* dkw    = (const float*)d_in[5];
  const float* skb    = (const float*)d_in[6];
  const float* dkb    = (const float*)d_in[7];
  const float* svw    = (const float*)d_in[8];
  const float* dvw    = (const float*)d_in[9];
  const float* svb    = (const float*)d_in[10];
  const float* dvb    = (const float*)d_in[11];
  const float* gamma  = (const float*)d_in[12];
  const float* beta   = (const float*)d_in[13];
  const int*   src    = (const int*)d_in[14];
  const int*   dst    = (const int*)d_in[15];

  const int N = in_sizes[0] / 32;   // 8000
  const int E = in_sizes[14];       // 40000

  float* out       = (float*)d_out;                  // N*32
  float* key_out   = out + (size_t)N * 32;           // E*32
  float* value_out = key_out + (size_t)E * 32;       // E*32
  float* attn_out  = value_out + (size_t)E * 32;     // E*4

  float* ws     = (float*)d_ws;
  float* logits = ws;                                // E*4 (later holds ex)
  float* mbuf   = logits + (size_t)E * 4;            // N*4
  float* denom  = mbuf + (size_t)N * 4;              // N*4
  float* agg    = denom + (size_t)N * 4;             // N*32

  init_kernel<<<(N * 32 + 255) / 256, 256, 0, stream>>>(mbuf, denom, agg, N);

  edge_kernel<<<(E + 7) / 8, 256, 0, stream>>>(feat, query, skw, dkw, skb, dkb,
                                               svw, dvw, svb, dvb, src, dst,
                                               key_out, value_out, logits, mbuf, E);

  expsum_kernel<<<(E * 4 + 255) / 256, 256, 0, stream>>>(logits, mbuf, denom, dst, E);

  attn_kernel<<<(E * 32 + 255) / 256, 256, 0, stream>>>(logits, denom, value_out, dst,
                                                        attn_out, agg, E);

  node_kernel<<<(N + 7) / 8, 256, 0, stream>>>(feat, node_w, node_b, gamma, beta,
                                               agg, out, N);
}